// EdgeAttention_37185826848829
// MI455X (gfx1250) — compile-verified
//
#include <hip/hip_runtime.h>
#include <cstddef>
#include <cstdint>

// ---------------------------------------------------------------------------
// Problem constants (from reference): N=192 nodes, NB=3 attrs, H=8 heads,
// D=10 in-feats, HU=[16,8] node widths, HE=[4,2] edge widths, Ne=N*N=36864.
// ---------------------------------------------------------------------------
constexpr int N_  = 192;
constexpr int NB_ = 3;
constexpr int H_  = 8;
constexpr int D_  = 10;
constexpr int HU0 = 16, HU1 = 8;
constexpr int HE0 = 4,  HE1 = 2;
constexpr int NE_ = N_ * N_;      // 36864
constexpr int OUTC = D_ + H_*HU0 + H_*HU1;  // 202

typedef float vf2 __attribute__((ext_vector_type(2)));
typedef float vf8 __attribute__((ext_vector_type(8)));

__device__ __forceinline__ float lrelu(float x) { return x > 0.f ? x : 0.2f * x; }

// ---------------------------------------------------------------------------
// fe[e][h*Fe+f] = sum_d X_edge[e][d]*we[h][d][f];  t[h][e] = fe . aen[h]
// layer0: X_edge[e][d] = E[d*Ne + e] (E.reshape(nb,-1).T), De=3, Fe=4
// layer1: X_edge = fe0 [Ne][32], De=32, Fe=2
// ---------------------------------------------------------------------------
__global__ __launch_bounds__(256) void fe_kernel(
    const float* __restrict__ xe, const float* __restrict__ we,
    const float* __restrict__ aen, float* __restrict__ fe,
    float* __restrict__ t, int De, int Fe, int layer)
{
  int e = blockIdx.x * 256 + threadIdx.x;
  if (e >= NE_) return;
  float x[32];
  if (layer == 0) {
#pragma unroll
    for (int d = 0; d < 3; ++d) x[d] = xe[(size_t)d * NE_ + e];
  } else {
    for (int d = 0; d < 32; ++d) x[d] = xe[(size_t)e * 32 + d];
  }
  const int W = H_ * Fe;
  for (int h = 0; h < H_; ++h) {
    float tv = 0.f;
    for (int f = 0; f < Fe; ++f) {
      float v = 0.f;
      for (int d = 0; d < De; ++d) v += x[d] * we[(h * De + d) * Fe + f];
      fe[(size_t)e * W + h * Fe + f] = v;
      tv += v * aen[h * Fe + f];
    }
    t[(size_t)h * NE_ + e] = tv;
  }
}

// ---------------------------------------------------------------------------
// feats[a][h][n][f] = sum_d X_head[n][d]*wn[a][h][d][f];  s = feats . aes[h]
// ---------------------------------------------------------------------------
__global__ __launch_bounds__(192) void feats_kernel(
    const float* __restrict__ xh, const float* __restrict__ wn,
    const float* __restrict__ aes, float* __restrict__ feats,
    float* __restrict__ s, int Din, int F, int layer)
{
  int n = threadIdx.x, h = blockIdx.x, a = blockIdx.y;
  const float* xr = (layer == 0) ? (xh + (size_t)n * Din)
                                 : (xh + ((size_t)a * N_ + n) * Din);
  const float* wr = wn + (size_t)(a * H_ + h) * Din * F;
  float sacc = 0.f;
  for (int f = 0; f < F; ++f) {
    float v = 0.f;
    for (int d = 0; d < Din; ++d) v += xr[d] * wr[(size_t)d * F + f];
    feats[(((size_t)a * H_ + h) * N_ + n) * F + f] = v;
    sacc += v * aes[h * F + f];
  }
  s[((size_t)a * H_ + h) * N_ + n] = sacc;
}

// ---------------------------------------------------------------------------
// Streaming edge->node attention (flash style, two-pass over Ne=36864):
// nfe[a][h][n][f] = softmax_e(lrelu(s+t)+adj[n]) @ fe[:, h*Fe+f] + be[h][f]
// All streams hit L2 (adj=28MB, t=1.2MB, fe<=4.7MB all L2-resident).
// ---------------------------------------------------------------------------
__device__ __forceinline__ float block_red_max(float v, float* red, int tid) {
  red[tid] = v; __syncthreads();
#pragma unroll
  for (int o = 128; o > 0; o >>= 1) {
    if (tid < o) red[tid] = fmaxf(red[tid], red[tid + o]);
    __syncthreads();
  }
  float r = red[0]; __syncthreads(); return r;
}
__device__ __forceinline__ float block_red_sum(float v, float* red, int tid) {
  red[tid] = v; __syncthreads();
#pragma unroll
  for (int o = 128; o > 0; o >>= 1) {
    if (tid < o) red[tid] += red[tid + o];
    __syncthreads();
  }
  float r = red[0]; __syncthreads(); return r;
}

__global__ __launch_bounds__(256) void edge_att_kernel(
    const float* __restrict__ s, const float* __restrict__ t,
    const float* __restrict__ adj, const float* __restrict__ fe,
    const float* __restrict__ be, float* __restrict__ nfe, int Fe)
{
  __shared__ float red[256];
  int tid = threadIdx.x;
  int n = blockIdx.x, h = blockIdx.y, a = blockIdx.z;
  float sval = s[((size_t)a * H_ + h) * N_ + n];
  const float* tr = t + (size_t)h * NE_;
  const float* ar = adj + (size_t)n * NE_;
  const int W = H_ * Fe;
  float m = -3.4e38f;
  for (int e = tid; e < NE_; e += 256)
    m = fmaxf(m, lrelu(sval + tr[e]) + ar[e]);
  float M = block_red_max(m, red, tid);
  float sum = 0.f, a0 = 0.f, a1 = 0.f, a2 = 0.f, a3 = 0.f;
  for (int e = tid; e < NE_; e += 256) {
    float p = __expf(lrelu(sval + tr[e]) + ar[e] - M);
    sum += p;
    const float* fr = fe + (size_t)e * W + h * Fe;
    a0 += p * fr[0];
    a1 += p * fr[1];
    if (Fe == 4) { a2 += p * fr[2]; a3 += p * fr[3]; }
  }
  sum = block_red_sum(sum, red, tid);
  a0 = block_red_sum(a0, red, tid);
  a1 = block_red_sum(a1, red, tid);
  if (Fe == 4) { a2 = block_red_sum(a2, red, tid); a3 = block_red_sum(a3, red, tid); }
  if (tid == 0) {
    float inv = 1.f / sum;
    float* o = nfe + (((size_t)a * H_ + h) * N_ + n) * Fe;
    o[0] = a0 * inv + be[h * Fe + 0];
    o[1] = a1 * inv + be[h * Fe + 1];
    if (Fe == 4) { o[2] = a2 * inv + be[h * Fe + 2]; o[3] = a3 * inv + be[h * Fe + 3]; }
  }
}

// ---------------------------------------------------------------------------
// f2 = concat(feats, nfe) @ wct + bct   (stored zero-padded to 16 cols for WMMA B)
// s2 = f2 . asf,  t2 = f2 . anf
// ---------------------------------------------------------------------------
__global__ __launch_bounds__(192) void f2_kernel(
    const float* __restrict__ feats, const float* __restrict__ nfe,
    const float* __restrict__ wct, const float* __restrict__ bct,
    const float* __restrict__ asf, const float* __restrict__ anf,
    float* __restrict__ f2, float* __restrict__ s2, float* __restrict__ t2,
    int F, int Fe)
{
  int n = threadIdx.x, h = blockIdx.x, a = blockIdx.y;
  const int Din = F + Fe;           // 20 (layer0) or 10 (layer1)
  float cat[20];
  const float* fr = feats + (((size_t)a * H_ + h) * N_ + n) * F;
  const float* nr = nfe   + (((size_t)a * H_ + h) * N_ + n) * Fe;
  for (int d = 0; d < F; ++d)  cat[d] = fr[d];
  for (int j = 0; j < Fe; ++j) cat[F + j] = nr[j];
  const float* wc = wct + (size_t)h * Din * F;
  float* f2r = f2 + (((size_t)a * H_ + h) * N_ + n) * 16;
  float sacc = 0.f, tacc = 0.f;
  for (int f = 0; f < 16; ++f) {
    float v = 0.f;
    if (f < F) {
      for (int d = 0; d < Din; ++d) v += cat[d] * wc[(size_t)d * F + f];
      v += bct[h * F + f];
      sacc += v * asf[((size_t)a * H_ + h) * F + f];
      tacc += v * anf[((size_t)a * H_ + h) * F + f];
    }
    f2r[f] = v;                     // zero pad cols f>=F
  }
  s2[((size_t)a * H_ + h) * N_ + n] = sacc;
  t2[((size_t)a * H_ + h) * N_ + n] = tacc;
}

// ---------------------------------------------------------------------------
// Rank-1 strength reduction of d2:
//   d2[h,n,k] = s2[h,n]*cs[k] + u[h,k],  cs[k]=colsum(E_info), u = t2 @ E_info
// ---------------------------------------------------------------------------
__global__ __launch_bounds__(192) void csu_kernel(
    const float* __restrict__ Ei_base, const float* __restrict__ t2,
    float* __restrict__ cs, float* __restrict__ u)
{
  __shared__ float t2l[H_][N_];
  int k = threadIdx.x, a = blockIdx.x;
  for (int h = 0; h < H_; ++h) t2l[h][k] = t2[((size_t)a * H_ + h) * N_ + k];
  __syncthreads();
  const float* Ei = Ei_base + (size_t)a * N_ * N_;
  float c = 0.f;
  float uu[H_] = {};
  for (int m = 0; m < N_; ++m) {
    float ev = Ei[(size_t)m * N_ + k];
    c += ev;
#pragma unroll
    for (int h = 0; h < H_; ++h) uu[h] += t2l[h][m] * ev;
  }
  cs[a * N_ + k] = c;
  for (int h = 0; h < H_; ++h) u[((size_t)a * H_ + h) * N_ + k] = uu[h];
}

// ---------------------------------------------------------------------------
// att2 softmax (per 16-row tile, into LDS) + nf = att2 @ f2 via
// V_WMMA_F32_16X16X4_F32 (exact fp32), then bias + ELU -> h1 head-concat.
// One wave (32 threads) per (attr, head, 16-row tile). EXEC all-ones at WMMA.
//
// B-matrix (f2, 12KB) is staged LDS via GLOBAL_LOAD_ASYNC_TO_LDS_B128 issued
// at kernel entry; the transfer overlaps the ~1100-VALU softmax phase and is
// fenced with a single s_wait_asynccnt 0 before the GEMM.  Inner loop is then
// pure LDS + WMMA.
//
// A-frag (16x4 f32): lanes 0-15 -> M=lane, K0/K1; lanes 16-31 -> M=lane-16, K2/K3.
// B-frag (4x16 f32): mirrored with N=lane&15.
// C/D: VGPR j -> M = j + 8*(lane>=16), N = lane&15.
// ---------------------------------------------------------------------------
__global__ __launch_bounds__(32) void att2_nf_kernel(
    const float* __restrict__ s2, const float* __restrict__ cs,
    const float* __restrict__ u,  const float* __restrict__ A,
    const float* __restrict__ f2, const float* __restrict__ bn,
    float* __restrict__ h1, float* __restrict__ einfo,
    int F, int writeE)
{
  __shared__ float att[16][196];   // stride 196 (196%64==4) -> conflict-free col reads
  __shared__ __align__(16) float f2s[N_ * 16];   // 12KB B-matrix stage
  int lane = threadIdx.x;
  int ntile = blockIdx.x, h = blockIdx.y, a = blockIdx.z;
  int n0 = ntile * 16;
  bool doE = (writeE != 0) && (h == H_ - 1);   // E_info = last head's att2

  // ---- kick off async global->LDS staging of this head's f2 [192x16] ----
  {
    const char* gsrc = (const char*)(f2 + ((size_t)a * H_ + h) * N_ * 16);
    uint32_t lds0 = (uint32_t)(uintptr_t)(&f2s[0]);   // low 32 bits = LDS offset
#pragma unroll
    for (int i = 0; i < (N_ * 16 * 4) / (32 * 16); ++i) {   // 24 x b128 per lane
      uint32_t off = (uint32_t)(lane + 32 * i) * 16u;
      uint32_t laddr = lds0 + off;
      uint64_t gaddr = (uint64_t)(uintptr_t)(gsrc + off);
      asm volatile("global_load_async_to_lds_b128 %0, %1, off"
                   :: "v"(laddr), "v"(gaddr) : "memory");
    }
  }

  const float* csr = cs + a * N_;
  const float* ur  = u + ((size_t)a * H_ + h) * N_;
  const float* Ar  = A + (size_t)a * N_ * N_;

  for (int r = 0; r < 16; ++r) {
    int n = n0 + r;
    float s2n = s2[((size_t)a * H_ + h) * N_ + n];
    float vals[6];
    float vmax = -3.4e38f;
#pragma unroll
    for (int j = 0; j < 6; ++j) {
      int k = lane + 32 * j;
      float v = lrelu(s2n * csr[k] + ur[k]) + Ar[(size_t)n * N_ + k];
      vals[j] = v;
      vmax = fmaxf(vmax, v);
    }
#pragma unroll
    for (int o = 16; o > 0; o >>= 1) vmax = fmaxf(vmax, __shfl_xor(vmax, o, 32));
    float lsum = 0.f;
#pragma unroll
    for (int j = 0; j < 6; ++j) { vals[j] = __expf(vals[j] - vmax); lsum += vals[j]; }
#pragma unroll
    for (int o = 16; o > 0; o >>= 1) lsum += __shfl_xor(lsum, o, 32);
    float inv = 1.f / lsum;
#pragma unroll
    for (int j = 0; j < 6; ++j) {
      int k = lane + 32 * j;
      float p = vals[j] * inv;
      att[r][k] = p;
      if (doE) einfo[((size_t)a * N_ + n) * N_ + k] = p;
    }
  }
  __syncthreads();
  asm volatile("s_wait_asynccnt 0x0" ::: "memory");   // f2s staging complete

  vf8 acc = {0.f, 0.f, 0.f, 0.f, 0.f, 0.f, 0.f, 0.f};
  int half = lane >> 4, l16 = lane & 15;
  for (int k0 = 0; k0 < N_; k0 += 4) {
    int kb = k0 + 2 * half;
    vf2 af, bf;
    af.x = att[l16][kb];
    af.y = att[l16][kb + 1];
    bf.x = f2s[kb * 16 + l16];
    bf.y = f2s[(kb + 1) * 16 + l16];
    acc = __builtin_amdgcn_wmma_f32_16x16x4_f32(
        false, af, false, bf, (short)0, acc, false, false);
  }
  int rofs = half * 8;
  float bias = (l16 < F) ? bn[((size_t)a * H_ + h) * F + l16] : 0.f;
#pragma unroll
  for (int j = 0; j < 8; ++j) {
    int n = n0 + rofs + j;
    float v = acc[j] + bias;
    v = v > 0.f ? v : (__expf(v) - 1.f);   // ELU
    if (l16 < F) h1[((size_t)a * N_ + n) * (H_ * F) + h * F + l16] = v;
  }
}

// ---------------------------------------------------------------------------
// out[n][a][0:10]=X, [10:138]=h1_layer0, [138:202]=h1_layer1
// ---------------------------------------------------------------------------
__global__ __launch_bounds__(224) void assemble_kernel(
    const float* __restrict__ X, const float* __restrict__ h10,
    const float* __restrict__ h11, float* __restrict__ out)
{
  int c = threadIdx.x, a = blockIdx.x, n = blockIdx.y;
  if (c >= OUTC) return;
  float v;
  if (c < D_)             v = X[(size_t)n * D_ + c];
  else if (c < D_ + 128)  v = h10[((size_t)a * N_ + n) * 128 + (c - D_)];
  else                    v = h11[((size_t)a * N_ + n) * 64 + (c - D_ - 128)];
  out[((size_t)n * NB_ + a) * OUTC + c] = v;
}

// ---------------------------------------------------------------------------
extern "C" void kernel_launch(void* const* d_in, const int* in_sizes, int n_in,
                              void* d_out, int out_size, void* d_ws, size_t ws_size,
                              hipStream_t stream)
{
  (void)in_sizes; (void)n_in; (void)out_size; (void)ws_size;
  const float* X     = (const float*)d_in[0];
  const float* A     = (const float*)d_in[1];
  const float* E     = (const float*)d_in[2];
  const float* adj   = (const float*)d_in[3];
  const float* w_n0  = (const float*)d_in[4];
  const float* b_n0  = (const float*)d_in[5];
  const float* as0   = (const float*)d_in[6];
  const float* an0   = (const float*)d_in[7];
  const float* w_n1  = (const float*)d_in[8];
  const float* b_n1  = (const float*)d_in[9];
  const float* as1   = (const float*)d_in[10];
  const float* an1   = (const float*)d_in[11];
  const float* w_e0  = (const float*)d_in[12];
  const float* b_e0  = (const float*)d_in[13];
  const float* w_ct0 = (const float*)d_in[14];
  const float* b_ct0 = (const float*)d_in[15];
  const float* aes0  = (const float*)d_in[16];
  const float* aen0  = (const float*)d_in[17];
  const float* w_e1  = (const float*)d_in[18];
  const float* b_e1  = (const float*)d_in[19];
  const float* w_ct1 = (const float*)d_in[20];
  const float* b_ct1 = (const float*)d_in[21];
  const float* aes1  = (const float*)d_in[22];
  const float* aen1  = (const float*)d_in[23];
  float* out = (float*)d_out;

  // workspace layout (floats), total ~11.6 MB
  float* w = (float*)d_ws;
  auto alloc = [&](size_t nf) { float* p = w; w += nf; return p; };
  float* fe0   = alloc((size_t)NE_ * H_ * HE0);
  float* fe1   = alloc((size_t)NE_ * H_ * HE1);
  float* t0    = alloc((size_t)H_ * NE_);
  float* t1    = alloc((size_t)H_ * NE_);
  float* feats0= alloc((size_t)NB_ * H_ * N_ * HU0);
  float* feats1= alloc((size_t)NB_ * H_ * N_ * HU1);
  float* s0    = alloc((size_t)NB_ * H_ * N_);
  float* s1    = alloc((size_t)NB_ * H_ * N_);
  float* nfe0  = alloc((size_t)NB_ * H_ * N_ * HE0);
  float* nfe1  = alloc((size_t)NB_ * H_ * N_ * HE1);
  float* f2_0  = alloc((size_t)NB_ * H_ * N_ * 16);
  float* f2_1  = alloc((size_t)NB_ * H_ * N_ * 16);
  float* s2_0  = alloc((size_t)NB_ * H_ * N_);
  float* t2_0  = alloc((size_t)NB_ * H_ * N_);
  float* s2_1  = alloc((size_t)NB_ * H_ * N_);
  float* t2_1  = alloc((size_t)NB_ * H_ * N_);
  float* cs0   = alloc((size_t)NB_ * N_);
  float* cs1   = alloc((size_t)NB_ * N_);
  float* u0    = alloc((size_t)NB_ * H_ * N_);
  float* u1    = alloc((size_t)NB_ * H_ * N_);
  float* einfo = alloc((size_t)NB_ * N_ * N_);
  float* h1_0  = alloc((size_t)NB_ * N_ * H_ * HU0);
  float* h1_1  = alloc((size_t)NB_ * N_ * H_ * HU1);

  const int feBlocks = (NE_ + 255) / 256;   // 144

  // ---------------- layer 0 ----------------
  fe_kernel<<<feBlocks, 256, 0, stream>>>(E, w_e0, aen0, fe0, t0, 3, HE0, 0);
  feats_kernel<<<dim3(H_, NB_), 192, 0, stream>>>(X, w_n0, aes0, feats0, s0, D_, HU0, 0);
  edge_att_kernel<<<dim3(N_, H_, NB_), 256, 0, stream>>>(s0, t0, adj, fe0, b_e0, nfe0, HE0);
  f2_kernel<<<dim3(H_, NB_), 192, 0, stream>>>(feats0, nfe0, w_ct0, b_ct0, as0, an0,
                                               f2_0, s2_0, t2_0, HU0, HE0);
  csu_kernel<<<NB_, 192, 0, stream>>>(E, t2_0, cs0, u0);
  att2_nf_kernel<<<dim3(N_ / 16, H_, NB_), 32, 0, stream>>>(
      s2_0, cs0, u0, A, f2_0, b_n0, h1_0, einfo, HU0, 1);

  // ---------------- layer 1 ----------------
  fe_kernel<<<feBlocks, 256, 0, stream>>>(fe0, w_e1, aen1, fe1, t1, H_ * HE0, HE1, 1);
  feats_kernel<<<dim3(H_, NB_), 192, 0, stream>>>(h1_0, w_n1, aes1, feats1, s1, H_ * HU0, HU1, 1);
  edge_att_kernel<<<dim3(N_, H_, NB_), 256, 0, stream>>>(s1, t1, adj, fe1, b_e1, nfe1, HE1);
  f2_kernel<<<dim3(H_, NB_), 192, 0, stream>>>(feats1, nfe1, w_ct1, b_ct1, as1, an1,
                                               f2_1, s2_1, t2_1, HU1, HE1);
  csu_kernel<<<NB_, 192, 0, stream>>>(einfo, t2_1, cs1, u1);
  att2_nf_kernel<<<dim3(N_ / 16, H_, NB_), 32, 0, stream>>>(
      s2_1, cs1, u1, A, f2_1, b_n1, h1_1, nullptr, HU1, 0);

  // ---------------- output ----------------
  assemble_kernel<<<dim3(NB_, N_), 224, 0, stream>>>(X, h1_0, h1_1, out);
}